// GCNFeaturePropagator_42580305772849
// MI455X (gfx1250) — compile-verified
//
#include <hip/hip_runtime.h>
#include <hip/hip_bf16.h>

// ---------------------------------------------------------------------------
// 2-layer GCN (PyG GCNConv semantics) for MI455X / gfx1250.
//   layer: deg -> dinv -> norm ; h = x@W (WMMA f32, W staged via TDM) ;
//          agg = selfloop-init + edge scatter ; +bias ; relu / sigmoid
// ---------------------------------------------------------------------------

typedef float v2f __attribute__((ext_vector_type(2)));
typedef float v8f __attribute__((ext_vector_type(8)));
typedef unsigned v4u __attribute__((ext_vector_type(4)));
typedef int v8i __attribute__((ext_vector_type(8)));
typedef int v4i __attribute__((ext_vector_type(4)));

#define FDIM 128

// ---------- degree / normalization ----------

__global__ void k_deg_init(float* deg, int n) {
    int i = blockIdx.x * blockDim.x + threadIdx.x;
    if (i < n) deg[i] = 1.0f;               // self-loop weight
}

__global__ void k_deg_accum(const long long* __restrict__ dst,
                            const float* __restrict__ ew,
                            float* __restrict__ deg, int e) {
    int i = blockIdx.x * blockDim.x + threadIdx.x;
    if (i < e) atomicAdd(&deg[(int)dst[i]], ew[i]);
}

__global__ void k_deg_to_dinv(float* deg, int n) {
    int i = blockIdx.x * blockDim.x + threadIdx.x;
    if (i < n) {
        float d = deg[i];
        deg[i] = (d > 0.0f) ? __frsqrt_rn(d) : 0.0f;
    }
}

__global__ void k_norm(const long long* __restrict__ src,
                       const long long* __restrict__ dst,
                       const float* __restrict__ ew,
                       const float* __restrict__ dinv,
                       float* __restrict__ norm, int e) {
    int i = blockIdx.x * blockDim.x + threadIdx.x;
    if (i < e) norm[i] = dinv[(int)src[i]] * ew[i] * dinv[(int)dst[i]];
}

// ---------- dense GEMM via V_WMMA_F32_16X16X4_F32 ----------
// Y[M x 128] = X[M x 128] @ W[128 x 128].
// grid.x = ceil(M/16); block = 256 (8 waves); wave w owns output cols [16w,16w+16).
// W (64 KB) staged into LDS once per workgroup by the Tensor Data Mover.

__global__ __launch_bounds__(256)
void k_gemm_wmma(const float* __restrict__ X, const float* __restrict__ W,
                 float* __restrict__ Y, int M) {
    __shared__ float lw[FDIM * FDIM];       // 64 KB of the 320 KB WGP LDS

    const int t = threadIdx.x;

#if __has_builtin(__builtin_amdgcn_tensor_load_to_lds) && \
    __has_builtin(__builtin_amdgcn_s_wait_tensorcnt)
    // --- TDM path: one wave issues a single 2D tensor DMA (ISA 08 §8.3/8.4) ---
    if (t < 32) {
        unsigned lds_off = (unsigned)(unsigned long long)(uintptr_t)lw; // LDS byte offset (addr[31:0])
        unsigned long long ga = (unsigned long long)(uintptr_t)W;
        v4u g0;
        g0[0] = 1u;                                           // count=1, user D#
        g0[1] = lds_off;                                      // lds_addr
        g0[2] = (unsigned)ga;                                 // global_addr[31:0]
        g0[3] = (unsigned)((ga >> 32) & 0x01FFFFFFu)          // global_addr[56:32]
              | (2u << 30);                                   // type = 2 ("image")
        v8i g1;
        g1[0] = (int)(2u << 16);        // workgroup_mask=0, data_size=2 (4 bytes)
        g1[1] = (int)(128u << 16);      // tensor_dim0 = 128 (low 16 in [31:16])
        g1[2] = (int)(128u << 16);      // tensor_dim0 hi=0 | tensor_dim1 = 128
        g1[3] = (int)(128u << 16);      // tensor_dim1 hi=0 | tile_dim0 = 128
        g1[4] = 128;                    // tile_dim1 = 128, tile_dim2 = 0
        g1[5] = 128;                    // tensor_dim0_stride = 128
        g1[6] = 0;                      // stride0 hi | tensor_dim1_stride lo
        g1[7] = 0;                      // tensor_dim1_stride hi
        v4i z4 = {0, 0, 0, 0};          // groups 2/3 unused (2D tensor)
        v8i z8 = {0, 0, 0, 0, 0, 0, 0, 0};
        __builtin_amdgcn_tensor_load_to_lds(g0, g1, z4, z4, z8, 0);
        __builtin_amdgcn_s_wait_tensorcnt(0);                 // TENSORcnt == 0
    }
    __syncthreads();
#else
    // --- fallback: cooperative vector load ---
    {
        const float4* wv = (const float4*)W;
        float4* lv = (float4*)lw;
        for (int i = t; i < FDIM * FDIM / 4; i += 256) lv[i] = wv[i];
        __syncthreads();
    }
#endif

    const int wave  = t >> 5;
    const int lane  = t & 31;
    const int m     = lane & 15;            // row-in-tile (A) / col-in-tile (B,C)
    const int khalf = lane >> 4;            // 0: K={0,1}, 1: K={2,3}
    const int row0  = blockIdx.x * 16;
    const int col0  = wave * 16;

    int arow = row0 + m;
    if (arow >= M) arow = M - 1;            // clamp (EXEC must stay all-1 for WMMA)
    const v2f* xv = (const v2f*)(X + (size_t)arow * FDIM);

    v8f acc = {};
    #pragma unroll 8
    for (int k0 = 0; k0 < FDIM; k0 += 4) {
        // A fragment: x[arow][k0 + 2*khalf + {0,1}]
        v2f a = xv[(k0 >> 1) + khalf];
        // B fragment: W[k0 + 2*khalf + {0,1}][col0 + m]
        v2f b;
        b.x = lw[(k0 + 2 * khalf + 0) * FDIM + col0 + m];
        b.y = lw[(k0 + 2 * khalf + 1) * FDIM + col0 + m];
        acc = __builtin_amdgcn_wmma_f32_16x16x4_f32(
            /*neg_a=*/false, a, /*neg_b=*/false, b,
            /*c_mod=*/(short)0, acc, /*reuse_a=*/false, /*reuse_b=*/false);
    }

    // D layout: VGPR r -> row (row0 + 8*khalf + r), col (col0 + m)
    float* y = Y + (size_t)(row0 + khalf * 8) * FDIM + col0 + m;
    if (row0 + 16 <= M) {                   // full tile: wave-uniform fast path
        #pragma unroll
        for (int r = 0; r < 8; ++r) y[(size_t)r * FDIM] = acc[r];
    } else {
        #pragma unroll
        for (int r = 0; r < 8; ++r)
            if (row0 + khalf * 8 + r < M) y[(size_t)r * FDIM] = acc[r];
    }
}

// ---------- aggregation ----------

// agg[v][f] = h[v][f] * dinv[v]^2   (self-loop term; also zero-inits poisoned buf)
__global__ void k_self_init(const float* __restrict__ H,
                            const float* __restrict__ dinv,
                            float* __restrict__ AGG, long long total) {
    long long i = (long long)blockIdx.x * blockDim.x + threadIdx.x;
    if (i < total) {
        float dv = dinv[(int)(i >> 7)];
        AGG[i] = H[i] * dv * dv;
    }
}

// one thread per (edge, 4-feature chunk): agg[dst] += h[src] * norm[e]
__global__ void k_edge_scatter(const long long* __restrict__ src,
                               const long long* __restrict__ dst,
                               const float* __restrict__ norm,
                               const float* __restrict__ H,
                               float* __restrict__ AGG, int e) {
    long long gid = (long long)blockIdx.x * blockDim.x + threadIdx.x;
    if (gid >= (long long)e * (FDIM / 4)) return;
    int ei = (int)(gid >> 5);               // FDIM/4 == 32 chunks per edge
    int ch = (int)(gid & 31);
    int s  = (int)src[ei];
    int d  = (int)dst[ei];
    float nr = norm[ei];
    const float4 hv = *(const float4*)(H + (size_t)s * FDIM + ch * 4);
    float* ag = AGG + (size_t)d * FDIM + ch * 4;
    atomicAdd(ag + 0, hv.x * nr);
    atomicAdd(ag + 1, hv.y * nr);
    atomicAdd(ag + 2, hv.z * nr);
    atomicAdd(ag + 3, hv.w * nr);
}

// ---------- epilogues ----------

__global__ void k_bias_relu(float* __restrict__ A, const float* __restrict__ b,
                            long long total) {
    long long i = (long long)blockIdx.x * blockDim.x + threadIdx.x;
    if (i < total) A[i] = fmaxf(A[i] + b[i & (FDIM - 1)], 0.0f);
}

__global__ void k_bias_sigmoid(float* __restrict__ A, const float* __restrict__ b,
                               long long total) {
    long long i = (long long)blockIdx.x * blockDim.x + threadIdx.x;
    if (i < total) {
        float v = A[i] + b[i & (FDIM - 1)];
        A[i] = 1.0f / (1.0f + __expf(-v));
    }
}

// ---------------------------------------------------------------------------

extern "C" void kernel_launch(void* const* d_in, const int* in_sizes, int n_in,
                              void* d_out, int out_size, void* d_ws, size_t ws_size,
                              hipStream_t stream) {
    const float*     x   = (const float*)d_in[0];
    const long long* ei  = (const long long*)d_in[1];     // [2, E] int64
    const float*     ew  = (const float*)d_in[2];
    const float*     W1  = (const float*)d_in[3];
    const float*     b1  = (const float*)d_in[4];
    const float*     W2  = (const float*)d_in[5];
    const float*     b2  = (const float*)d_in[6];

    const int N = in_sizes[0] / FDIM;
    const int E = in_sizes[2];
    const long long* src = ei;
    const long long* dst = ei + E;

    // workspace carve-up (all sizes are multiples of 16 bytes)
    float* dinv = (float*)d_ws;                              // N
    float* nrm  = dinv + N;                                  // E
    float* bufH = nrm + E;                                   // N*128 (gemm out)
    float* bufA = bufH + (size_t)N * FDIM;                   // N*128 (agg / relu)
    float* out  = (float*)d_out;

    const long long NF = (long long)N * FDIM;
    const int TB = 256;
    dim3 blk(TB);
    dim3 g_n((N + TB - 1) / TB);
    dim3 g_e((E + TB - 1) / TB);
    dim3 g_nf((unsigned)((NF + TB - 1) / TB));
    dim3 g_sc((unsigned)(((long long)E * (FDIM / 4) + TB - 1) / TB));
    dim3 g_mm((N + 15) / 16);

    // normalization (shared by both layers)
    k_deg_init   <<<g_n,  blk, 0, stream>>>(dinv, N);
    k_deg_accum  <<<g_e,  blk, 0, stream>>>(dst, ew, dinv, E);
    k_deg_to_dinv<<<g_n,  blk, 0, stream>>>(dinv, N);
    k_norm       <<<g_e,  blk, 0, stream>>>(src, dst, ew, dinv, nrm, E);

    // layer 1
    k_gemm_wmma   <<<g_mm, blk, 0, stream>>>(x, W1, bufH, N);
    k_self_init   <<<g_nf, blk, 0, stream>>>(bufH, dinv, bufA, NF);
    k_edge_scatter<<<g_sc, blk, 0, stream>>>(src, dst, nrm, bufH, bufA, E);
    k_bias_relu   <<<g_nf, blk, 0, stream>>>(bufA, b1, NF);

    // layer 2 (aggregate directly into d_out)
    k_gemm_wmma   <<<g_mm, blk, 0, stream>>>(bufA, W2, bufH, N);
    k_self_init   <<<g_nf, blk, 0, stream>>>(bufH, dinv, out, NF);
    k_edge_scatter<<<g_sc, blk, 0, stream>>>(src, dst, nrm, bufH, out, E);
    k_bias_sigmoid<<<g_nf, blk, 0, stream>>>(out, b2, NF);
}